// Model_61710090109310
// MI455X (gfx1250) — compile-verified
//
#include <hip/hip_runtime.h>
#include <math.h>

#define N_NODES 200000
#define N_EDGES 2000000
#define IN_DIM  165
#define HID_DIM 128
#define KPAD    168   // IN_DIM padded up to multiple of 4 (zero fill)
#define KMAIN   164   // main K loop bound: kb = k + kp (kp<=2) stays < IN_DIM for k < 164

typedef __attribute__((ext_vector_type(2))) float v2f;
typedef __attribute__((ext_vector_type(8))) float v8f;

// ---------------- degree / normalization ----------------

__global__ void k_deg_init(float* __restrict__ deg, int n) {
    int i = blockIdx.x * blockDim.x + threadIdx.x;
    if (i < n) deg[i] = 1.0f;   // self-loop contributes 1
}

__global__ void k_deg_count(const int* __restrict__ dst, float* __restrict__ deg, int e) {
    int i = blockIdx.x * blockDim.x + threadIdx.x;
    if (i < e) atomicAdd(&deg[dst[i]], 1.0f);
}

__global__ void k_dinv(float* __restrict__ deg, int n) {
    int i = blockIdx.x * blockDim.x + threadIdx.x;
    if (i < n) deg[i] = rsqrtf(deg[i]);   // deg >= 1 always (self-loops)
}

// ---------------- layer-1 GEMM: h = x @ W1^T  (fp32 WMMA 16x16x4) ----------------
// Block: 256 threads = 8 waves, handles 64 nodes x 128 features.
// Wave w: node tile = w>>1 (16 nodes), feature half = (w&1)*64, loops 4 feature tiles.
// Main K loop is guard-free; single masked tail step handles k=164.

__global__ __launch_bounds__(256)
void k_gemm1(const float* __restrict__ x, const float* __restrict__ W1,
             float* __restrict__ h) {
    __shared__ float sX[64][KPAD];   // 43008 bytes
    const int tid      = threadIdx.x;
    const int nodeBase = blockIdx.x * 64;

    // Stage 64 rows of x into LDS, zero-padded to KPAD columns (coalesced).
    for (int idx = tid; idx < 64 * KPAD; idx += 256) {
        int r = idx / KPAD, c = idx % KPAD;
        sX[r][c] = (c < IN_DIM) ? x[(nodeBase + r) * IN_DIM + c] : 0.0f;
    }
    __syncthreads();

    const int lane  = tid & 31;
    const int wave  = tid >> 5;
    const int m     = lane & 15;            // A row / B col / C col within tile
    const int kp    = (lane >> 4) << 1;     // 0 or 2: K sub-offset for A/B fragments
    const int ntile = wave >> 1;            // 0..3 -> 16-node tile
    const int fbase = (wave & 1) * 64;      // feature half
    const float* arow = &sX[ntile * 16 + m][0];

    for (int ft = 0; ft < 4; ++ft) {
        const int feat0 = fbase + ft * 16;
        v8f c = {0.f, 0.f, 0.f, 0.f, 0.f, 0.f, 0.f, 0.f};
        const float* wrow = &W1[(feat0 + m) * IN_DIM];   // B col m == W1 row feat0+m

        // Guard-free main loop: k = 0..160 step 4; kb+1 <= 163 < IN_DIM always.
        for (int k = 0; k < KMAIN; k += 4) {
            v2f a, b;
            a.x = arow[k + kp];          // A[m][k+kp]     (LDS)
            a.y = arow[k + kp + 1];      // A[m][k+kp+1]
            b.x = wrow[k + kp];          // B[k+kp][m] = W1[feat0+m][k+kp]
            b.y = wrow[k + kp + 1];
            c = __builtin_amdgcn_wmma_f32_16x16x4_f32(
                    false, a, false, b, (short)0, c, false, false);
        }
        // Tail step k=164: only kb==164 (kp==0 half) is in range -> branch-free select.
        {
            v2f a, b;
            a.x = arow[KMAIN + kp];      // LDS zero-padded past IN_DIM
            a.y = arow[KMAIN + kp + 1];
            b.x = (kp == 0) ? wrow[KMAIN] : 0.0f;   // W1[.,164]; 165..167 are zero
            b.y = 0.0f;
            c = __builtin_amdgcn_wmma_f32_16x16x4_f32(
                    false, a, false, b, (short)0, c, false, false);
        }

        // C layout: VGPR r -> M = r + 8*(lane>>4), N = lane&15
        const int mBase = ntile * 16 + ((lane >> 4) << 3);
#pragma unroll
        for (int r = 0; r < 8; ++r) {
            h[(nodeBase + mBase + r) * HID_DIM + feat0 + m] = c[r];
        }
    }
}

// ---------------- layer-1 aggregation ----------------
// Self-loop term also initializes agg (no separate memset).

__global__ void k_selfloop1(const float* __restrict__ h, const float* __restrict__ dinv,
                            float* __restrict__ agg, int n) {
    int gid = blockIdx.x * blockDim.x + threadIdx.x;   // n*32 threads
    int i = gid >> 5;
    if (i >= n) return;
    int f4 = (gid & 31) * 4;
    float w = dinv[i] * dinv[i];
    float4 hv = *(const float4*)&h[(size_t)i * HID_DIM + f4];
    float4 o;
    o.x = hv.x * w; o.y = hv.y * w; o.z = hv.z * w; o.w = hv.w * w;
    *(float4*)&agg[(size_t)i * HID_DIM + f4] = o;
}

// One wave per edge; each lane moves 4 features (float4 gather, 4 atomics).
__global__ void k_scatter1(const int* __restrict__ src, const int* __restrict__ dst,
                           const float* __restrict__ dinv, const float* __restrict__ h,
                           float* __restrict__ agg) {
    int gid = blockIdx.x * blockDim.x + threadIdx.x;   // N_EDGES*32 threads
    int e = gid >> 5;
    if (e >= N_EDGES) return;
    int lane = threadIdx.x & 31;
    int s = src[e], d = dst[e];
    float w = dinv[s] * dinv[d];
    int f4 = lane * 4;
    float4 hv = *(const float4*)&h[(size_t)s * HID_DIM + f4];
    float* ap = &agg[(size_t)d * HID_DIM + f4];
    atomicAdd(ap + 0, hv.x * w);
    atomicAdd(ap + 1, hv.y * w);
    atomicAdd(ap + 2, hv.z * w);
    atomicAdd(ap + 3, hv.w * w);
}

// ---------------- fuse relu + bias + dot(W2): per-node scalar ----------------

__global__ void k_relu_dot(const float* __restrict__ agg, const float* __restrict__ b1,
                           const float* __restrict__ W2, float* __restrict__ sval, int n) {
    int gid = blockIdx.x * blockDim.x + threadIdx.x;   // n*32 threads, wave per node
    int i = gid >> 5;
    if (i >= n) return;
    int lane = threadIdx.x & 31;
    int f4 = lane * 4;
    float4 a  = *(const float4*)&agg[(size_t)i * HID_DIM + f4];
    float4 bb = *(const float4*)&b1[f4];
    float4 w2 = *(const float4*)&W2[f4];
    float p = fmaxf(a.x + bb.x, 0.f) * w2.x
            + fmaxf(a.y + bb.y, 0.f) * w2.y
            + fmaxf(a.z + bb.z, 0.f) * w2.z
            + fmaxf(a.w + bb.w, 0.f) * w2.w;
#pragma unroll
    for (int off = 16; off > 0; off >>= 1)
        p += __shfl_xor(p, off, 32);
    if (lane == 0) sval[i] = p;
}

// ---------------- layer-2 scalar aggregation ----------------

__global__ void k_selfloop2(const float* __restrict__ sval, const float* __restrict__ dinv,
                            float* __restrict__ out, int n) {
    int i = blockIdx.x * blockDim.x + threadIdx.x;
    if (i < n) out[i] = sval[i] * dinv[i] * dinv[i];   // initializes d_out
}

__global__ void k_scatter2(const int* __restrict__ src, const int* __restrict__ dst,
                           const float* __restrict__ dinv, const float* __restrict__ sval,
                           float* __restrict__ out) {
    int e = blockIdx.x * blockDim.x + threadIdx.x;
    if (e >= N_EDGES) return;
    int s = src[e], d = dst[e];
    atomicAdd(&out[d], sval[s] * dinv[s] * dinv[d]);
}

__global__ void k_sigmoid(float* __restrict__ out, const float* __restrict__ b2, int n) {
    int i = blockIdx.x * blockDim.x + threadIdx.x;
    if (i < n) out[i] = 1.0f / (1.0f + expf(-(out[i] + b2[0])));
}

// ---------------- launcher ----------------

extern "C" void kernel_launch(void* const* d_in, const int* in_sizes, int n_in,
                              void* d_out, int out_size, void* d_ws, size_t ws_size,
                              hipStream_t stream) {
    (void)in_sizes; (void)n_in; (void)out_size; (void)ws_size;
    const float* x  = (const float*)d_in[0];
    const float* W1 = (const float*)d_in[1];
    const float* b1 = (const float*)d_in[2];
    const float* W2 = (const float*)d_in[3];
    const float* b2 = (const float*)d_in[4];
    const int*   ei = (const int*)d_in[5];      // int32 (JAX x64 disabled)
    const int* src = ei;                        // edge_index[0]
    const int* dst = ei + N_EDGES;              // edge_index[1]
    float* out = (float*)d_out;

    float* dinv = (float*)d_ws;                         // N floats (deg -> dinv)
    float* sval = dinv + N_NODES;                       // N floats
    float* h    = sval + N_NODES;                       // N*128 floats
    float* agg  = h + (size_t)N_NODES * HID_DIM;        // N*128 floats

    const int B = 256;
    k_deg_init <<<(N_NODES + B - 1) / B, B, 0, stream>>>(dinv, N_NODES);
    k_deg_count<<<(N_EDGES + B - 1) / B, B, 0, stream>>>(dst, dinv, N_EDGES);
    k_dinv     <<<(N_NODES + B - 1) / B, B, 0, stream>>>(dinv, N_NODES);

    k_gemm1    <<<N_NODES / 64, B, 0, stream>>>(x, W1, h);

    k_selfloop1<<<(N_NODES * 32) / B, B, 0, stream>>>(h, dinv, agg, N_NODES);
    k_scatter1 <<<((size_t)N_EDGES * 32) / B, B, 0, stream>>>(src, dst, dinv, h, agg);

    k_relu_dot <<<(N_NODES * 32) / B, B, 0, stream>>>(agg, b1, W2, sval, N_NODES);

    k_selfloop2<<<(N_NODES + B - 1) / B, B, 0, stream>>>(sval, dinv, out, N_NODES);
    k_scatter2 <<<(N_EDGES + B - 1) / B, B, 0, stream>>>(src, dst, dinv, sval, out);
    k_sigmoid  <<<(N_NODES + B - 1) / B, B, 0, stream>>>(out, b2, N_NODES);
}